// ReEigLayer_26147760898374
// MI455X (gfx1250) — compile-verified
//
#include <hip/hip_runtime.h>
#include <math.h>
#include <stdint.h>

// ReEig: f(X) = U max(L, tau) U^T  for symmetric X.
// Matrix-function form: f(X) = 0.5*X + 0.5*sign(X - tau I)*(X - tau I) + 0.5*tau*I
// sign(Y) via Newton-Schulz: S <- 1.5 S - 0.5 S^3 (all GEMMs -> fp32 WMMA).
// Staging uses CDNA5 async global->LDS copies (ASYNCcnt) with double buffering.

#define NN 4096
#define TAU 1e-5f
#define NS_ITERS 30   // even: final sign matrix ends in workspace buffer
#define PI_ITERS 12   // power iterations for spectral-norm estimate

typedef float v2f __attribute__((ext_vector_type(2)));
typedef float v8f __attribute__((ext_vector_type(8)));

// GEMM tiling
#define BM 128
#define BN 128
#define BK 32
#define LDA_S 36    // BK + 4 pad (floats); 144B row stride, 16B aligned
#define LDB_S 132   // BN + 4 pad (floats); 528B row stride, 16B aligned

__device__ __forceinline__ void async_load_b128(uint32_t lds_off,
                                                const float* __restrict__ base,
                                                uint32_t byte_off) {
  // GVS form: vdst = LDS byte address, vaddr = 32-bit byte offset, saddr = base
  asm volatile("global_load_async_to_lds_b128 %0, %1, %2"
               :
               : "v"(lds_off), "v"(byte_off), "s"(base)
               : "memory");
}

// MODE 0: C = A*B
// MODE 1: C = 1.5*E - 0.5*(A*B)              (E = S at output location)
// MODE 2: C = 0.5*(E + A*(B - tau I)) + 0.5*tau*I   (E = X, B = X)
template <int MODE>
__global__ __launch_bounds__(256)
void gemm_ns(const float* __restrict__ A, const float* __restrict__ B,
             const float* __restrict__ E, float* __restrict__ C) {
  __shared__ float As[2][BM][LDA_S];
  __shared__ float Bs[2][BK][LDB_S];

  const int t    = threadIdx.x;
  const int lane = t & 31;
  const int wave = t >> 5;
  const int mBase = blockIdx.y * BM;
  const int nBase = blockIdx.x * BN;
  const int rowBase = (wave >> 1) * 32;  // 4 wave-rows
  const int colBase = (wave & 1) * 64;   // 2 wave-cols
  const int l15  = lane & 15;
  const int koff = (lane >> 4) * 2;      // K-pair select per ISA A/B fragment layout

  // LDS byte offsets (flat->LDS aperture keeps DS offset in addr[31:0])
  const uint32_t asBase = (uint32_t)(uintptr_t)&As[0][0][0];
  const uint32_t bsBase = (uint32_t)(uintptr_t)&Bs[0][0][0];
  const uint32_t asBufStride = (uint32_t)(BM * LDA_S * 4);
  const uint32_t bsBufStride = (uint32_t)(BK * LDB_S * 4);

  // Per-tile async staging: 8 B128 commands per thread (A:4, B:4) => ASYNCcnt 8/wave.
  auto issue_tile = [&](int kb, int buf) {
#pragma unroll
    for (int i = 0; i < 4; ++i) {  // A tile: BM x BK
      int idx = t + i * 256;
      int r   = idx >> 3;          // 0..127
      int kq  = (idx & 7) * 4;     // 0..28
      uint32_t goff = (uint32_t)((((size_t)(mBase + r)) * NN + (kb + kq)) * 4);
      uint32_t loff = asBase + (uint32_t)buf * asBufStride +
                      (uint32_t)((r * LDA_S + kq) * 4);
      async_load_b128(loff, A, goff);
    }
#pragma unroll
    for (int i = 0; i < 4; ++i) {  // B tile: BK x BN
      int idx = t + i * 256;
      int kk  = idx >> 5;          // 0..31
      int cq  = (idx & 31) * 4;    // 0..124
      uint32_t goff = (uint32_t)((((size_t)(kb + kk)) * NN + (nBase + cq)) * 4);
      uint32_t loff = bsBase + (uint32_t)buf * bsBufStride +
                      (uint32_t)((kk * LDB_S + cq) * 4);
      async_load_b128(loff, B, goff);
    }
  };

  v8f acc[2][4];
  const v8f vzero = {0.f, 0.f, 0.f, 0.f, 0.f, 0.f, 0.f, 0.f};
#pragma unroll
  for (int mi = 0; mi < 2; ++mi)
#pragma unroll
    for (int ni = 0; ni < 4; ++ni) acc[mi][ni] = vzero;

  issue_tile(0, 0);  // prologue

  int cur = 0;
  for (int kb = 0; kb < NN; kb += BK) {
    const bool hasNext = (kb + BK) < NN;
    if (hasNext) {
      issue_tile(kb + BK, cur ^ 1);
      // completion is in-order: <=8 outstanding => current tile's 8 are done
      asm volatile("s_wait_asynccnt 0x8" ::: "memory");
    } else {
      asm volatile("s_wait_asynccnt 0x0" ::: "memory");
    }
    __syncthreads();

    if (MODE == 2) {
      // Patch B tile in LDS: B := X - tau*I on the diagonal
      if (t < BK) {
        int c = kb + t - nBase;
        if (c >= 0 && c < BN) Bs[cur][t][c] -= TAU;
      }
      __syncthreads();
    }

#pragma unroll
    for (int kk = 0; kk < BK; kk += 4) {
      v2f a[2], b[4];
#pragma unroll
      for (int mi = 0; mi < 2; ++mi) {
        int r = rowBase + mi * 16 + l15;
        a[mi].x = As[cur][r][kk + koff];
        a[mi].y = As[cur][r][kk + koff + 1];
      }
#pragma unroll
      for (int ni = 0; ni < 4; ++ni) {
        int c = colBase + ni * 16 + l15;
        b[ni].x = Bs[cur][kk + koff][c];
        b[ni].y = Bs[cur][kk + koff + 1][c];
      }
#pragma unroll
      for (int mi = 0; mi < 2; ++mi)
#pragma unroll
        for (int ni = 0; ni < 4; ++ni)
          acc[mi][ni] = __builtin_amdgcn_wmma_f32_16x16x4_f32(
              false, a[mi], false, b[ni], (short)0, acc[mi][ni], false, false);
    }
    __syncthreads();  // all waves done reading buf[cur] before it is overwritten
    cur ^= 1;
  }

  // Epilogue. C/D layout: VGPR r -> M = r + 8*(lane>>4), N = lane&15.
  const int mSub = 8 * (lane >> 4);
#pragma unroll
  for (int mi = 0; mi < 2; ++mi) {
#pragma unroll
    for (int ni = 0; ni < 4; ++ni) {
      int n = nBase + colBase + ni * 16 + l15;
#pragma unroll
      for (int r = 0; r < 8; ++r) {
        int m = mBase + rowBase + mi * 16 + r + mSub;
        size_t off = (size_t)m * NN + n;
        float v = acc[mi][ni][r];
        if (MODE == 0) {
          C[off] = v;
        } else if (MODE == 1) {
          C[off] = 1.5f * E[off] - 0.5f * v;
        } else {
          float o = 0.5f * (E[off] + v);
          if (m == n) o += 0.5f * TAU;
          C[off] = o;
        }
      }
    }
  }
}

__global__ __launch_bounds__(256)
void fill_ones(float* __restrict__ x) {
  int i = blockIdx.x * 256 + threadIdx.x;
  if (i < NN) x[i] = 1.0f / 64.0f;
}

// y = X * x  (one block per row)
__global__ __launch_bounds__(256)
void gemv_k(const float* __restrict__ M, const float* __restrict__ x,
            float* __restrict__ y) {
  __shared__ float red[256];
  int row = blockIdx.x;
  const float* Mr = M + (size_t)row * NN;
  float s = 0.f;
  for (int k = threadIdx.x; k < NN; k += 256) s += Mr[k] * x[k];
  red[threadIdx.x] = s;
  __syncthreads();
  for (int w = 128; w > 0; w >>= 1) {
    if (threadIdx.x < w) red[threadIdx.x] += red[threadIdx.x + w];
    __syncthreads();
  }
  if (threadIdx.x == 0) y[row] = red[0];
}

__global__ __launch_bounds__(256)
void norm_k(const float* __restrict__ x, float* __restrict__ out) {
  __shared__ float red[256];
  float s = 0.f;
  for (int k = threadIdx.x; k < NN; k += 256) s += x[k] * x[k];
  red[threadIdx.x] = s;
  __syncthreads();
  for (int w = 128; w > 0; w >>= 1) {
    if (threadIdx.x < w) red[threadIdx.x] += red[threadIdx.x + w];
    __syncthreads();
  }
  if (threadIdx.x == 0) *out = sqrtf(red[0]);
}

// S0 = (X - tau*I) / (1.1 * sigma_est),  sigma_est = nrm[1]/nrm[0]
__global__ __launch_bounds__(256)
void init_s0(const float* __restrict__ X, const float* __restrict__ nrm,
             float* __restrict__ S) {
  float sigma = nrm[1] / fmaxf(nrm[0], 1e-30f);
  float s = 1.0f / (1.10f * fmaxf(sigma, 1e-20f));
  size_t i = (size_t)blockIdx.x * 256 + threadIdx.x;  // grid covers NN*NN exactly
  int row = (int)(i >> 12);
  int col = (int)(i & (NN - 1));
  float v = X[i];
  if (row == col) v -= TAU;
  S[i] = s * v;
}

extern "C" void kernel_launch(void* const* d_in, const int* in_sizes, int n_in,
                              void* d_out, int out_size, void* d_ws, size_t ws_size,
                              hipStream_t stream) {
  (void)in_sizes; (void)n_in; (void)out_size; (void)ws_size;
  const float* X = (const float*)d_in[0];
  float* OUT = (float*)d_out;

  // Workspace layout (floats): W0 [N*N] | W1 [N*N] | vx [N] | vy [N] | nrm [2]
  float* W0 = (float*)d_ws;
  float* W1 = W0 + (size_t)NN * NN;
  float* vx = W1 + (size_t)NN * NN;
  float* vy = vx + NN;
  float* nrm = vy + NN;

  // --- spectral norm estimate via power iteration ---
  fill_ones<<<NN / 256, 256, 0, stream>>>(vx);
  for (int i = 0; i < PI_ITERS; ++i) {
    gemv_k<<<NN, 256, 0, stream>>>(X, vx, vy);
    float* tmp = vx; vx = vy; vy = tmp;
  }
  // vx = x_k, vy = x_{k-1}
  norm_k<<<1, 256, 0, stream>>>(vy, nrm + 0);
  norm_k<<<1, 256, 0, stream>>>(vx, nrm + 1);

  // --- S0 = (X - tau I) * scale ---
  init_s0<<<(NN * (size_t)NN) / 256, 256, 0, stream>>>(X, nrm, W0);

  // --- Newton-Schulz sign iteration: S <- 1.5 S - 0.5 S^3 ---
  dim3 grid(NN / BN, NN / BM);
  float* S = W0;
  float* T = OUT;  // ping-pong partner; NS_ITERS even => final S lands in W0
  for (int it = 0; it < NS_ITERS; ++it) {
    gemm_ns<0><<<grid, 256, 0, stream>>>(S, S, S, W1);   // Z = S*S
    gemm_ns<1><<<grid, 256, 0, stream>>>(S, W1, S, T);   // T = 1.5S - 0.5 S*Z
    float* tmp = S; S = T; T = tmp;
  }

  // --- OUT = 0.5*(X + sign(Y)*(X - tau I)) + 0.5*tau*I ---
  gemm_ns<2><<<grid, 256, 0, stream>>>(S, X, X, OUT);
}